// GCN_32109175505054
// MI455X (gfx1250) — compile-verified
//
#include <hip/hip_runtime.h>
#include <hip/hip_bf16.h>

// CDNA5 (gfx1250) wave32 WMMA types
typedef __attribute__((ext_vector_type(2))) float v2f;
typedef __attribute__((ext_vector_type(8))) float v8f;

// ---------------------------------------------------------------------------
// Utility kernels
// ---------------------------------------------------------------------------
__global__ __launch_bounds__(256) void gcn_zero_kernel(float4* __restrict__ p, int n4) {
    int t = blockIdx.x * blockDim.x + threadIdx.x;
    if (t < n4) p[t] = make_float4(0.f, 0.f, 0.f, 0.f);
}

__global__ __launch_bounds__(256) void gcn_degree_kernel(const int* __restrict__ src,
                                                         const int* __restrict__ dst,
                                                         float* __restrict__ deg_out,
                                                         float* __restrict__ deg_in, int E) {
    int e = blockIdx.x * blockDim.x + threadIdx.x;
    if (e >= E) return;
    atomicAdd(&deg_out[src[e]], 1.0f);
    atomicAdd(&deg_in[dst[e]], 1.0f);
}

__global__ __launch_bounds__(256) void gcn_rsqrt_kernel(float* __restrict__ d, int n) {
    int t = blockIdx.x * blockDim.x + threadIdx.x;
    if (t < n) d[t] = rsqrtf(fmaxf(d[t], 1.0f));
}

// ---------------------------------------------------------------------------
// GEMM1: H1[n,64] = (feat[n,0:256] * norm_out[n]) @ W1[256,64]  (f32 WMMA)
// One wave -> one 16x16 output tile via V_WMMA_F32_16X16X4_F32, K=256.
// Two accumulators (even/odd K steps) break the WMMA->WMMA RAW chain.
// W1 staged transposed in LDS so A and B fragments are contiguous v2f loads:
//   lane l: M/N = l&15, K pair = k + 2*(l>>4) .. +1   (ISA 16x4 f32 layout)
// ---------------------------------------------------------------------------
__global__ __launch_bounds__(256) void gcn_gemm1_wmma(const float* __restrict__ feat,
                                                      const float* __restrict__ norm_out,
                                                      const float* __restrict__ W1,
                                                      float* __restrict__ H1, int nrows) {
    __shared__ float Wt[64 * 256]; // 64 KB of the 320 KB WGP LDS
    for (int i = threadIdx.x; i < 256 * 64; i += 256) {
        int k = i >> 6, n = i & 63;
        Wt[n * 256 + k] = W1[i];     // transpose: Wt[n][k]
    }
    __syncthreads();

    const int wave = threadIdx.x >> 5;
    const int lane = threadIdx.x & 31;
    const int mtiles = (nrows + 15) >> 4;
    const int mt = blockIdx.x * 2 + (wave >> 2);   // wave-uniform -> EXEC stays all-1s
    if (mt >= mtiles) return;
    const int nt = wave & 3;                        // 4 N-tiles cover 64 cols

    const int m0   = mt << 4;
    const int row  = m0 + (lane & 15);
    const int rowc = row < nrows ? row : nrows - 1; // clamp for safe loads
    const int koff = (lane >> 4) << 1;              // 0 or 2

    const float* __restrict__ arow = feat + (size_t)rowc * 256;
    const float  scale             = norm_out[rowc];
    const float* __restrict__ bcol = Wt + ((nt << 4) + (lane & 15)) * 256;

    v8f acc0 = {};
    v8f acc1 = {};
#pragma unroll 4
    for (int k = 0; k < 256; k += 8) {
        v2f a0 = *(const v2f*)(arow + k + koff);
        v2f a1 = *(const v2f*)(arow + k + 4 + koff);
        a0 = a0 * scale;                             // fold norm_out into A
        a1 = a1 * scale;
        v2f b0 = *(const v2f*)(bcol + k + koff);
        v2f b1 = *(const v2f*)(bcol + k + 4 + koff);
        acc0 = __builtin_amdgcn_wmma_f32_16x16x4_f32(false, a0, false, b0,
                                                     (short)0, acc0, false, false);
        acc1 = __builtin_amdgcn_wmma_f32_16x16x4_f32(false, a1, false, b1,
                                                     (short)0, acc1, false, false);
    }
    v8f acc = acc0 + acc1;

    // C/D layout: VGPR r -> row r (lanes 0-15) / r+8 (lanes 16-31), col = lane&15
    const int col   = (nt << 4) + (lane & 15);
    const int rbase = m0 + ((lane >> 4) << 3);
    float* __restrict__ o = H1 + (size_t)rbase * 64 + col;
    if (m0 + 16 <= nrows) {            // wave-uniform fast path: no exec juggling
#pragma unroll
        for (int r = 0; r < 8; ++r) o[(size_t)r * 64] = acc[r];
    } else {                           // ragged tail tile only
#pragma unroll
        for (int r = 0; r < 8; ++r)
            if (rbase + r < nrows) o[(size_t)r * 64] = acc[r];
    }
}

// ---------------------------------------------------------------------------
// Edge aggregation, layer 1: agg[dst] += H1[src], 64 f32/edge.
// 16 threads per edge, float4 gather + 4 f32 global atomics each.
// H1 (12.8 MB) and agg1 (12.8 MB) both stay resident in the 192 MB L2.
// ---------------------------------------------------------------------------
__global__ __launch_bounds__(256) void gcn_agg64_kernel(const float* __restrict__ msg,
                                                        const int* __restrict__ src,
                                                        const int* __restrict__ dst,
                                                        float* __restrict__ agg, int E) {
    int t = blockIdx.x * blockDim.x + threadIdx.x;
    int e = t >> 4;
    if (e >= E) return;
    int c = (t & 15) << 2;
    int s = src[e], d = dst[e];
    const float4 v = *(const float4*)(msg + (size_t)s * 64 + c);
    float* a = agg + (size_t)d * 64 + c;
    atomicAdd(a + 0, v.x);
    atomicAdd(a + 1, v.y);
    atomicAdd(a + 2, v.z);
    atomicAdd(a + 3, v.w);
}

// Epilogue 1: H1post = relu(agg1*norm_in + b1) * norm_out   (pre-scale layer 2 input)
__global__ __launch_bounds__(256) void gcn_finish1_kernel(const float* __restrict__ agg1,
                                                          const float* __restrict__ norm_in,
                                                          const float* __restrict__ norm_out,
                                                          const float* __restrict__ b1,
                                                          float* __restrict__ H1post, int n) {
    int t = blockIdx.x * blockDim.x + threadIdx.x;
    if (t >= n * 64) return;
    int node = t >> 6, c = t & 63;
    float v = fmaf(agg1[t], norm_in[node], b1[c]);
    v = fmaxf(v, 0.0f) * norm_out[node];
    H1post[t] = v;
}

// ---------------------------------------------------------------------------
// GEMM2: T2[n,48] = H1post[n,0:64] @ W2pad[64,48]   (cols 40..47 exact zeros)
// One wave -> one 16-row stripe, loops the 3 N-tiles. f32 WMMA, K=64,
// dual accumulators per tile.
// ---------------------------------------------------------------------------
__global__ __launch_bounds__(256) void gcn_gemm2_wmma(const float* __restrict__ H1post,
                                                      const float* __restrict__ W2,
                                                      float* __restrict__ T2, int nrows) {
    __shared__ float Wt[48 * 64]; // 12 KB, transposed + zero padded: Wt[n][k]
    for (int i = threadIdx.x; i < 48 * 64; i += 256) {
        int n = i >> 6, k = i & 63;
        Wt[i] = (n < 40) ? W2[k * 40 + n] : 0.0f;
    }
    __syncthreads();

    const int wave = threadIdx.x >> 5;
    const int lane = threadIdx.x & 31;
    const int mtiles = (nrows + 15) >> 4;
    const int mt = blockIdx.x * 8 + wave;           // wave-uniform guard
    if (mt >= mtiles) return;

    const int m0   = mt << 4;
    const int row  = m0 + (lane & 15);
    const int rowc = row < nrows ? row : nrows - 1;
    const int koff = (lane >> 4) << 1;
    const float* __restrict__ arow = H1post + (size_t)rowc * 64;
    const bool full = (m0 + 16 <= nrows);

#pragma unroll
    for (int nt = 0; nt < 3; ++nt) {
        const float* __restrict__ bcol = Wt + ((nt << 4) + (lane & 15)) * 64;
        v8f acc0 = {};
        v8f acc1 = {};
#pragma unroll
        for (int k = 0; k < 64; k += 8) {
            v2f a0 = *(const v2f*)(arow + k + koff);
            v2f a1 = *(const v2f*)(arow + k + 4 + koff);
            v2f b0 = *(const v2f*)(bcol + k + koff);
            v2f b1 = *(const v2f*)(bcol + k + 4 + koff);
            acc0 = __builtin_amdgcn_wmma_f32_16x16x4_f32(false, a0, false, b0,
                                                         (short)0, acc0, false, false);
            acc1 = __builtin_amdgcn_wmma_f32_16x16x4_f32(false, a1, false, b1,
                                                         (short)0, acc1, false, false);
        }
        v8f acc = acc0 + acc1;

        const int col   = (nt << 4) + (lane & 15);
        const int rbase = m0 + ((lane >> 4) << 3);
        float* __restrict__ o = T2 + (size_t)rbase * 48 + col;
        if (full) {
#pragma unroll
            for (int r = 0; r < 8; ++r) o[(size_t)r * 48] = acc[r];
        } else {
#pragma unroll
            for (int r = 0; r < 8; ++r)
                if (rbase + r < nrows) o[(size_t)r * 48] = acc[r];
        }
    }
}

// Edge aggregation, layer 2: 48-padded rows, 12 of 16 threads per edge active.
__global__ __launch_bounds__(256) void gcn_agg48_kernel(const float* __restrict__ msg,
                                                        const int* __restrict__ src,
                                                        const int* __restrict__ dst,
                                                        float* __restrict__ agg, int E) {
    int t = blockIdx.x * blockDim.x + threadIdx.x;
    int e = t >> 4;
    if (e >= E) return;
    int j = t & 15;
    if (j >= 12) return;
    int c = j << 2;
    int s = src[e], d = dst[e];
    const float4 v = *(const float4*)(msg + (size_t)s * 48 + c);
    float* a = agg + (size_t)d * 48 + c;
    atomicAdd(a + 0, v.x);
    atomicAdd(a + 1, v.y);
    atomicAdd(a + 2, v.z);
    atomicAdd(a + 3, v.w);
}

// Epilogue 2: out[n,40] = agg2[n,0:40]*norm_in + b2
__global__ __launch_bounds__(256) void gcn_finish2_kernel(const float* __restrict__ agg2,
                                                          const float* __restrict__ norm_in,
                                                          const float* __restrict__ b2,
                                                          float* __restrict__ out, int n) {
    int t = blockIdx.x * blockDim.x + threadIdx.x;
    if (t >= n * 40) return;
    int node = t / 40, c = t - node * 40;
    out[t] = fmaf(agg2[(size_t)node * 48 + c], norm_in[node], b2[c]);
}

// ---------------------------------------------------------------------------
// Launch
// ---------------------------------------------------------------------------
extern "C" void kernel_launch(void* const* d_in, const int* in_sizes, int n_in,
                              void* d_out, int out_size, void* d_ws, size_t ws_size,
                              hipStream_t stream) {
    const float* feat = (const float*)d_in[0];
    const int*   src  = (const int*)d_in[1];
    const int*   dst  = (const int*)d_in[2];
    const float* W1   = (const float*)d_in[3];
    const float* b1   = (const float*)d_in[4];
    const float* W2   = (const float*)d_in[5];
    const float* b2   = (const float*)d_in[6];

    const int n = in_sizes[0] / 256;  // 50000 nodes
    const int E = in_sizes[1];        // 800000 edges

    const int nA = (n + 3) & ~3;      // float4-aligned node count
    float* ws       = (float*)d_ws;
    float* norm_out = ws;                       // [nA]
    float* norm_in  = ws + nA;                  // [nA] (contiguous with norm_out)
    float* H1       = norm_in + nA;             // [n*64]  raw gemm1 out, then relu'd/scaled
    float* agg1     = H1 + (size_t)n * 64;      // [n*64]
    float* T2       = agg1 + (size_t)n * 64;    // [n*48]
    float* agg2     = T2 + (size_t)n * 48;      // [n*48]

    const int B = 256;
    const int mtiles = (n + 15) >> 4;

    // Zero: degrees (both norm arrays) + both aggregation buffers (every launch).
    {
        int c4 = (2 * nA) >> 2;
        gcn_zero_kernel<<<(c4 + B - 1) / B, B, 0, stream>>>((float4*)norm_out, c4);
        c4 = (n * 64) >> 2;
        gcn_zero_kernel<<<(c4 + B - 1) / B, B, 0, stream>>>((float4*)agg1, c4);
        c4 = (n * 48) >> 2;
        gcn_zero_kernel<<<(c4 + B - 1) / B, B, 0, stream>>>((float4*)agg2, c4);
    }

    // Degrees -> norms
    gcn_degree_kernel<<<(E + B - 1) / B, B, 0, stream>>>(src, dst, norm_out, norm_in, E);
    gcn_rsqrt_kernel<<<(2 * nA + B - 1) / B, B, 0, stream>>>(norm_out, 2 * nA);

    // Layer 1
    gcn_gemm1_wmma<<<(mtiles + 1) / 2, B, 0, stream>>>(feat, norm_out, W1, H1, n);
    {
        long long t = (long long)E * 16;
        gcn_agg64_kernel<<<(int)((t + B - 1) / B), B, 0, stream>>>(H1, src, dst, agg1, E);
    }
    gcn_finish1_kernel<<<(n * 64 + B - 1) / B, B, 0, stream>>>(agg1, norm_in, norm_out, b1, H1, n);

    // Layer 2
    gcn_gemm2_wmma<<<(mtiles + 7) / 8, B, 0, stream>>>(H1, W2, T2, n);
    {
        long long t = (long long)E * 16;
        gcn_agg48_kernel<<<(int)((t + B - 1) / B), B, 0, stream>>>(T2, src, dst, agg2, E);
    }
    gcn_finish2_kernel<<<(n * 40 + B - 1) / B, B, 0, stream>>>(agg2, norm_in, b2, (float*)d_out, n);
}